// TwistToMatrix_88536455839887
// MI455X (gfx1250) — compile-verified
//
#include <hip/hip_runtime.h>

typedef float v2f __attribute__((ext_vector_type(2)));
typedef float v4f __attribute__((ext_vector_type(4)));

// ---------------------------------------------------------------------------
// One SE(3) exponential. Mathematically identical to the reference:
//   A = [[W, t],[0,0]], W = hat(w),  E = I + A + c2*A^2 + c3*A^3
// using W^2 = w w^T - th2*I and W^3 = -th2*W:
//   R = I + (1 - c3*th2)*W + c2*(w w^T - th2*I)
//   p = t + c2*(w x t) + c3*(w*(w.t) - th2*t)
// Small-angle branch polynomials match the reference's jnp.where exactly.
// Cost trimmed for the 23.3 TB/s roofline: one sincosf (shared range
// reduction) + ONE division per row: inv = 1/th2s, c2 = (1-c)*inv,
// c3 = (th-s)*th*inv*inv   (th*inv^2 == 1/(th2s*th) exactly).
// ---------------------------------------------------------------------------
__device__ __forceinline__ void se3_exp_one(float* __restrict__ o,
                                            float wx, float wy, float wz,
                                            float tx, float ty, float tz) {
    float th2   = wx * wx + wy * wy + wz * wz;
    bool  small = th2 < 1e-8f;
    float th2s  = small ? 1.0f : th2;
    float th    = sqrtf(th2s);
    float s, c;
    sincosf(th, &s, &c);
    float inv   = 1.0f / th2s;                    // the only division
    float c2    = small ? (0.5f - th2 * (1.0f / 24.0f))
                        : (1.0f - c) * inv;
    float c3    = small ? ((1.0f / 6.0f) - th2 * (1.0f / 120.0f))
                        : (th - s) * th * inv * inv;
    float s1    = 1.0f - c3 * th2;   // == sin(th)/th
    float c2t   = c2 * th2;

    float xx = c2 * wx * wx, yy = c2 * wy * wy, zz = c2 * wz * wz;
    float xy = c2 * wx * wy, xz = c2 * wx * wz, yz = c2 * wy * wz;

    float r00 = 1.0f - c2t + xx;
    float r01 = xy - s1 * wz;
    float r02 = xz + s1 * wy;
    float r10 = xy + s1 * wz;
    float r11 = 1.0f - c2t + yy;
    float r12 = yz - s1 * wx;
    float r20 = xz - s1 * wy;
    float r21 = yz + s1 * wx;
    float r22 = 1.0f - c2t + zz;

    float cx = wy * tz - wz * ty;
    float cy = wz * tx - wx * tz;
    float cz = wx * ty - wy * tx;
    float wt = wx * tx + wy * ty + wz * tz;
    float p0 = tx + c2 * cx + c3 * (wx * wt - th2 * tx);
    float p1 = ty + c2 * cy + c3 * (wy * wt - th2 * ty);
    float p2 = tz + c2 * cz + c3 * (wz * wt - th2 * tz);

    v4f r0 = {r00, r01, r02, p0};
    v4f r1 = {r10, r11, r12, p1};
    v4f r2 = {r20, r21, r22, p2};
    v4f r3 = {0.0f, 0.0f, 0.0f, 1.0f};

    v4f* po = (v4f*)o;                       // 16B-aligned (row base = 64*i)
    __builtin_nontemporal_store(r0, po + 0); // streamed: written once, never re-read
    __builtin_nontemporal_store(r1, po + 1);
    __builtin_nontemporal_store(r2, po + 2);
    __builtin_nontemporal_store(r3, po + 3);
}

// ---------------------------------------------------------------------------
// 256 threads/block = 8 wave32; 2 rows/thread -> 512 rows/block.
// Full blocks stage their input through LDS with the gfx1250 async DMA path:
//   each wave owns a 1536 B LDS span (64 rows * 24 B); each lane issues 3
//   GLOBAL_LOAD_ASYNC_TO_LDS_B128 (16B-aligned, perfectly coalesced), then
//   s_wait_asynccnt 0. A wave reads only data it loaded itself, so no
//   workgroup barrier is required and the wait latency matches a direct load.
// Partial blocks (generic safety; N=2M has none) use direct NT vector loads.
// ---------------------------------------------------------------------------
__global__ __launch_bounds__(256) void se3_exp_kernel(
    const float* __restrict__ tw, float* __restrict__ out, int n_items) {

    __shared__ alignas(16) float sbuf[512 * 6];   // 12 KB (LDS is 320 KB/WGP)

    const int  tid  = threadIdx.x;
    const int  lane = tid & 31;
    const int  wave = tid >> 5;
    const long blockItemBase = (long)blockIdx.x * 512;
    const long i0 = blockItemBase + (long)tid * 2;

    if (blockItemBase + 512 <= (long)n_items) {
        // ---- full block: async-DMA the wave's input span into LDS ----
        const char* gwave =
            (const char*)(tw + (blockItemBase + (long)wave * 64) * 6); // 16B aligned
        unsigned lwave = (unsigned)(size_t)&sbuf[wave * 64 * 6];       // LDS byte offset
#pragma unroll
        for (int k = 0; k < 3; ++k) {
            unsigned    loff = lwave + (unsigned)(lane + 32 * k) * 16u;
            const char* g    = gwave + (lane + 32 * k) * 16;
            asm volatile("global_load_async_to_lds_b128 %0, %1, off"
                         :: "v"(loff), "v"(g) : "memory");
        }
        // Speculative stream prefetch ~96KB ahead (global_prefetch_b8; OOB
        // speculative prefetches are silently dropped per ISA 10.5).
        __builtin_prefetch(gwave + 96 * 1024, 0, 0);

        asm volatile("s_wait_asynccnt 0x0" ::: "memory");

        const v4f* ls = (const v4f*)&sbuf[(wave * 64 + lane * 2) * 6]; // 48B stride
        v4f q0 = ls[0];
        v4f q1 = ls[1];
        v4f q2 = ls[2];
        se3_exp_one(out + i0 * 16,       q0[0], q0[1], q0[2], q0[3], q1[0], q1[1]);
        se3_exp_one(out + (i0 + 1) * 16, q1[2], q1[3], q2[0], q2[1], q2[2], q2[3]);
    } else {
        // ---- partial block: direct NT vector loads, per-row guards ----
        if (i0 >= n_items) return;
        const float* base = tw + i0 * 6;
        if (i0 + 1 < n_items) {
            const v4f* p = (const v4f*)base;      // byte offset 48*t: 16B aligned
            v4f q0 = __builtin_nontemporal_load(p + 0);
            v4f q1 = __builtin_nontemporal_load(p + 1);
            v4f q2 = __builtin_nontemporal_load(p + 2);
            se3_exp_one(out + i0 * 16,       q0[0], q0[1], q0[2], q0[3], q1[0], q1[1]);
            se3_exp_one(out + (i0 + 1) * 16, q1[2], q1[3], q2[0], q2[1], q2[2], q2[3]);
        } else {
            const v2f* p = (const v2f*)base;      // 8B-aligned, stays in bounds
            v2f a = __builtin_nontemporal_load(p + 0);
            v2f b = __builtin_nontemporal_load(p + 1);
            v2f c = __builtin_nontemporal_load(p + 2);
            se3_exp_one(out + i0 * 16, a[0], a[1], b[0], b[1], c[0], c[1]);
        }
    }
}

extern "C" void kernel_launch(void* const* d_in, const int* in_sizes, int n_in,
                              void* d_out, int out_size, void* d_ws, size_t ws_size,
                              hipStream_t stream) {
    const float* twist = (const float*)d_in[0];
    float*       out   = (float*)d_out;
    const int n_items  = in_sizes[0] / 6;              // 2,097,152

    const int threads = 256;                           // 8 waves (wave32)
    const int rows_per_block = 512;                    // 2 rows/thread
    const int blocks = (int)(((long)n_items + rows_per_block - 1) / rows_per_block);

    se3_exp_kernel<<<blocks, threads, 0, stream>>>(twist, out, n_items);
}